// _Whitening_36876589203673
// MI455X (gfx1250) — compile-verified
//
#include <hip/hip_runtime.h>
#include <hip/hip_bf16.h>
#include <math.h>

// Whitening: x[32,512,56,56] f32, 32 groups of 16 channels.
// Pass 1: fused channel-sums + per-group Gram via V_WMMA_F32_16X16X4_F32,
//         b128 loads feeding 2 WMMAs each (Gram is slot-assignment invariant).
// Pass 2: tiny Cholesky + triangular inverse (32 threads).
// Pass 3: out = Linv @ (x - mean) via V_WMMA_F32_16X16X4_F32, NT stores.

typedef __attribute__((ext_vector_type(2))) float v2f;
typedef __attribute__((ext_vector_type(4))) float v4f;
typedef __attribute__((ext_vector_type(8))) float v8f;

#define BATCH 32
#define CHN   512
#define HW    3136      // 56*56
#define GS    16
#define NG    32        // CHN / GS
#define EPSF  1.0e-3f

// ---------------------------------------------------------------------------
// Pass 1: per (b, g) block of 8 waves. Each wave owns 392 positions of the
// 3136-position plane set: 49 iterations x (1 b128 load -> 2 WMMA rank-4
// Gram updates) + per-channel sums. Reduce via LDS ds_add_f32 + global atomics.
// WMMA f32 16x16x4 layouts (ISA 7.12.2):
//   A (16x4):  lane<16: M=lane, v0:K=0 v1:K=1 ; lane>=16: M=lane-16, v0:K=2 v1:K=3
//   B (4x16):  mirrored with N=lane%16 -> same register image as A (B = A^T data)
//   D (16x16): vgpr v: M = v + 8*(lane>=16), N = lane%16
// Position->slot mapping per iteration i (8 positions):
//   WMMA#1: half0 lanes {8i, 8i+1}, half1 lanes {8i+4, 8i+5}
//   WMMA#2: half0 lanes {8i+2,8i+3}, half1 lanes {8i+6, 8i+7}
// ---------------------------------------------------------------------------
__global__ void whiten_stats_kernel(const float* __restrict__ x,
                                    float* __restrict__ S,
                                    float* __restrict__ csum) {
    const int g    = blockIdx.x % NG;
    const int b    = blockIdx.x / NG;
    const int wave = threadIdx.x >> 5;
    const int lane = threadIdx.x & 31;
    const int half = lane >> 4;
    const int lm   = lane & 15;   // channel within group

    const float* lptr = x + ((long long)b * CHN + (long long)g * GS + lm) * HW
                          + wave * 392 + half * 4;

    v8f acc = {};
    float lsum = 0.0f;
    for (int it = 0; it < 49; ++it) {
        v4f q = *(const v4f*)lptr;    // 4 consecutive positions, 16B aligned
        lptr += 8;
        lsum += (q.x + q.y) + (q.z + q.w);
        v2f ab0; ab0.x = q.x; ab0.y = q.y;   // even-aligned sub-pair of quad
        v2f ab1; ab1.x = q.z; ab1.y = q.w;
        acc = __builtin_amdgcn_wmma_f32_16x16x4_f32(
                  false, ab0, false, ab0, (short)0, acc, false, false);
        acc = __builtin_amdgcn_wmma_f32_16x16x4_f32(
                  false, ab1, false, ab1, (short)0, acc, false, false);
    }

    __shared__ float sS[256];
    __shared__ float sC[16];
    sS[threadIdx.x] = 0.0f;
    if (threadIdx.x < 16) sC[threadIdx.x] = 0.0f;
    __syncthreads();

    #pragma unroll
    for (int v = 0; v < 8; ++v) {
        const int row = v + half * 8;            // D-matrix M index
        atomicAdd(&sS[row * 16 + lm], acc[v]);   // ds_add_f32
    }
    atomicAdd(&sC[lm], lsum);
    __syncthreads();

    atomicAdd(&S[g * 256 + threadIdx.x], sS[threadIdx.x]);
    if (threadIdx.x < 16) atomicAdd(&csum[g * GS + threadIdx.x], sC[threadIdx.x]);
}

// ---------------------------------------------------------------------------
// Pass 2: one thread per group. cov = S/N - m m^T, shrink, Cholesky, invert L.
// ---------------------------------------------------------------------------
__global__ void whiten_factor_kernel(const float* __restrict__ S,
                                     const float* __restrict__ csum,
                                     float* __restrict__ Linv,
                                     float* __restrict__ mean) {
    const int g = threadIdx.x;
    if (g >= NG) return;
    const float invN = 1.0f / (float)((long long)BATCH * HW);

    float m[GS];
    for (int i = 0; i < GS; ++i) m[i] = csum[g * GS + i] * invN;

    float A[GS][GS];
    for (int i = 0; i < GS; ++i)
        for (int j = 0; j < GS; ++j) {
            float cov = S[g * 256 + i * GS + j] * invN - m[i] * m[j];
            A[i][j] = (1.0f - EPSF) * cov + (i == j ? EPSF : 0.0f);
        }

    // In-place Cholesky (lower).
    for (int k = 0; k < GS; ++k) {
        float d = sqrtf(A[k][k]);
        A[k][k] = d;
        const float inv = 1.0f / d;
        for (int i = k + 1; i < GS; ++i) A[i][k] *= inv;
        for (int j = k + 1; j < GS; ++j)
            for (int i = j; i < GS; ++i)
                A[i][j] -= A[i][k] * A[j][k];
    }

    // Forward substitution: L * Y = I  ->  Y = Linv (lower triangular).
    float Y[GS][GS];
    for (int c = 0; c < GS; ++c) {
        for (int r = 0; r < GS; ++r) Y[r][c] = 0.0f;
        for (int r = c; r < GS; ++r) {
            float s = (r == c) ? 1.0f : 0.0f;
            for (int k = c; k < r; ++k) s -= A[r][k] * Y[k][c];
            Y[r][c] = s / A[r][r];
        }
    }

    for (int r = 0; r < GS; ++r)
        for (int c = 0; c < GS; ++c)
            Linv[g * 256 + r * GS + c] = Y[r][c];
    for (int i = 0; i < GS; ++i) mean[g * GS + i] = m[i];
}

// ---------------------------------------------------------------------------
// Pass 3: out(16 x 16cols) = Linv(16x16) @ (x - m), 4 chained K=4 WMMAs per
// 16-position tile. Each wave covers 7 consecutive tiles (112 positions);
// 28 waves cover one (b,g) plane-set of 3136 positions. Output stores are
// non-temporal so the write stream doesn't evict x from the 192MB L2.
// ---------------------------------------------------------------------------
__global__ void whiten_apply_kernel(const float* __restrict__ x,
                                    const float* __restrict__ Linv,
                                    const float* __restrict__ mean,
                                    float* __restrict__ out) {
    const int wave = threadIdx.x >> 5;
    const int lane = threadIdx.x & 31;
    const int half = lane >> 4;
    const int lm   = lane & 15;

    const int wgl  = blockIdx.x * (blockDim.x >> 5) + wave;
    const int bg   = wgl / 28;
    const int wseg = wgl % 28;
    const int b    = bg / NG;
    const int g    = bg % NG;

    const long long planeBase = ((long long)b * CHN + (long long)g * GS) * HW;
    const float* xb = x + planeBase;
    float* ob = out + planeBase;

    // A slices of Linv and matching per-channel means (8B contiguous pairs).
    v2f aA[4], mB[4];
    #pragma unroll
    for (int kc = 0; kc < 4; ++kc) {
        const int col = kc * 4 + half * 2;
        aA[kc] = *(const v2f*)&Linv[g * 256 + lm * GS + col];
        mB[kc] = *(const v2f*)&mean[g * GS + col];
    }

    const int pos0 = wseg * 112;
    for (int t = 0; t < 7; ++t) {
        const int p = pos0 + t * 16 + lm;   // this lane's column (N = lm)
        v8f acc = {};
        #pragma unroll
        for (int kc = 0; kc < 4; ++kc) {
            const int ch = kc * 4 + half * 2;   // B-layout K rows for this lane
            v2f bv;
            bv.x = xb[(long long)ch * HW + p]       - mB[kc].x;
            bv.y = xb[(long long)(ch + 1) * HW + p] - mB[kc].y;
            acc = __builtin_amdgcn_wmma_f32_16x16x4_f32(
                      false, aA[kc], false, bv, (short)0, acc, false, false);
        }
        #pragma unroll
        for (int v = 0; v < 8; ++v) {
            const int row = v + half * 8;        // D-matrix M index
            __builtin_nontemporal_store(acc[v], &ob[(long long)row * HW + p]);
        }
    }
}

// ---------------------------------------------------------------------------
extern "C" void kernel_launch(void* const* d_in, const int* in_sizes, int n_in,
                              void* d_out, int out_size, void* d_ws, size_t ws_size,
                              hipStream_t stream) {
    const float* x = (const float*)d_in[0];
    float* out = (float*)d_out;

    float* ws    = (float*)d_ws;
    float* S     = ws;            // NG*256 = 8192 floats
    float* csum  = ws + 8192;     // CHN    =  512 floats
    float* Linv  = ws + 8704;     // NG*256 = 8192 floats
    float* meanp = ws + 16896;    // CHN    =  512 floats

    // Zero the accumulators (S + csum). Graph-capture safe.
    hipMemsetAsync(S, 0, (size_t)(8192 + 512) * sizeof(float), stream);

    // Pass 1: 1024 blocks x 256 threads (8 waves), one block per (b, g).
    whiten_stats_kernel<<<dim3(NG * BATCH), dim3(256), 0, stream>>>(x, S, csum);

    // Pass 2: tiny.
    whiten_factor_kernel<<<dim3(1), dim3(32), 0, stream>>>(S, csum, Linv, meanp);

    // Pass 3: BATCH*NG*28 waves / 8 waves per block = 3584 blocks.
    whiten_apply_kernel<<<dim3((BATCH * NG * 28) / 8), dim3(256), 0, stream>>>(
        x, Linv, meanp, out);
}